// FaceZFwd_22290880266412
// MI455X (gfx1250) — compile-verified
//
#include <hip/hip_runtime.h>
#include <math.h>

// MI455X / gfx1250, wave32. One wave per frame; 8 frames per 256-thread block.
// Memory-bound (~152 MB traffic -> ~6.5 us at 23.3 TB/s); the per-frame
// rot_mat x positions einsum runs on V_WMMA_F32_16X16X4_F32.
// All cross-lane ops (ds_bpermute) execute unconditionally at full EXEC.

typedef __attribute__((ext_vector_type(2))) float v2f;
typedef __attribute__((ext_vector_type(8))) float v8f;

#define EPSF 1e-8f
#define PIF  3.14159265358979323846f

__device__ __forceinline__ float shfl32(float v, int src) {
    return __shfl(v, src, 32);
}

// rotmat (row-major, 9 floats) -> quaternion -> axis-angle (reference semantics)
__device__ __forceinline__ void mat_to_axis_angle(const float m[9], float aa[3]) {
    const float m00 = m[0], m01 = m[1], m02 = m[2];
    const float m10 = m[3], m11 = m[4], m12 = m[5];
    const float m20 = m[6], m21 = m[7], m22 = m[8];
    const float tr = m00 + m11 + m22;

    const float s0  = sqrtf(fmaxf(tr + 1.0f, EPSF)) * 2.0f;
    const float rs0 = 1.0f / s0;
    const float q0w = 0.25f * s0;
    const float q0x = (m21 - m12) * rs0;
    const float q0y = (m02 - m20) * rs0;
    const float q0z = (m10 - m01) * rs0;

    const float s1  = sqrtf(fmaxf(1.0f + m00 - m11 - m22, EPSF)) * 2.0f;
    const float rs1 = 1.0f / s1;
    const float q1w = (m21 - m12) * rs1;
    const float q1x = 0.25f * s1;
    const float q1y = (m01 + m10) * rs1;
    const float q1z = (m02 + m20) * rs1;

    const float s2  = sqrtf(fmaxf(1.0f - m00 + m11 - m22, EPSF)) * 2.0f;
    const float rs2 = 1.0f / s2;
    const float q2w = (m02 - m20) * rs2;
    const float q2x = (m01 + m10) * rs2;
    const float q2y = 0.25f * s2;
    const float q2z = (m12 + m21) * rs2;

    const float s3  = sqrtf(fmaxf(1.0f - m00 - m11 + m22, EPSF)) * 2.0f;
    const float rs3 = 1.0f / s3;
    const float q3w = (m10 - m01) * rs3;
    const float q3x = (m02 + m20) * rs3;
    const float q3y = (m12 + m21) * rs3;
    const float q3z = 0.25f * s3;

    const bool c0 = tr > 0.0f;
    const bool c1 = (m00 > m11) && (m00 > m22);
    const bool c2 = m11 > m22;

    const float w = c0 ? q0w : (c1 ? q1w : (c2 ? q2w : q3w));
    const float x = c0 ? q0x : (c1 ? q1x : (c2 ? q2x : q3x));
    const float y = c0 ? q0y : (c1 ? q1y : (c2 ? q2y : q3y));
    const float z = c0 ? q0z : (c1 ? q1z : (c2 ? q2z : q3z));

    const float sin_sq = x * x + y * y + z * z;
    const float sin_t  = sqrtf(fmaxf(sin_sq, EPSF));
    // sin_t > 0 strictly, so atan2(-sin_t, -w) == atan2(sin_t, w) - pi exactly.
    const float t1 = atan2f(sin_t, w);
    const float two_theta = 2.0f * ((w < 0.0f) ? (t1 - PIF) : t1);
    const float k = (sin_sq > EPSF) ? (two_theta / sin_t) : 2.0f;
    aa[0] = x * k;
    aa[1] = y * k;
    aa[2] = z * k;
}

__global__ __launch_bounds__(256)
void facez_kernel(const float* __restrict__ rot,      // (BF, 64, 3, 3)
                  const float* __restrict__ pos,      // (BF, 64, 3)
                  float* __restrict__ out_pos,        // (BF, 64, 3)
                  float* __restrict__ out_aa,         // (BF, 64, 3)
                  float* __restrict__ out_rm,         // (BF, 3, 3)
                  int nframes) {
    const int tid   = threadIdx.x;
    const int lane  = tid & 31;
    const int wave  = tid >> 5;
    const int frame = blockIdx.x * 8 + wave;
    if (frame >= nframes) return;  // wave-uniform branch: EXEC stays all-ones

    const float* pf = pos + (size_t)frame * 192;
    const float* rf = rot + (size_t)frame * 576;

    // Prefetch the rotation stream for this frame (global_prefetch_b8 path)
    __builtin_prefetch(rf + lane * 18, 0, 0);

    // ---- load two joints per lane (v0 = lane, v1 = lane + 32) ----
    const int v0 = lane, v1 = lane + 32;
    const float p0x = pf[v0 * 3 + 0], p0y = pf[v0 * 3 + 1], p0z = pf[v0 * 3 + 2];
    const float p1x = pf[v1 * 3 + 0], p1y = pf[v1 * 3 + 1], p1z = pf[v1 * 3 + 2];

    // ---- across = p[1] - p[2] + p[16] - p[17], normalized ----
    float ax = shfl32(p0x, 1) - shfl32(p0x, 2) + shfl32(p0x, 16) - shfl32(p0x, 17);
    float ay = shfl32(p0y, 1) - shfl32(p0y, 2) + shfl32(p0y, 16) - shfl32(p0y, 17);
    float az = shfl32(p0z, 1) - shfl32(p0z, 2) + shfl32(p0z, 16) - shfl32(p0z, 17);
    float inv = 1.0f / sqrtf(ax * ax + ay * ay + az * az);
    ax *= inv; ay *= inv; az *= inv;

    // fwd = normalize(cross(across, up=(0,1,0))) = normalize((-az, 0, ax))
    float fx = -az, fy = 0.0f, fz = ax;
    inv = 1.0f / sqrtf(fx * fx + fy * fy + fz * fz);
    fx *= inv; fy *= inv; fz *= inv;

    // q = normalize([1 + fz, cross(fwd, (0,0,1))]) = normalize([1+fz, fy, -fx, 0])
    float qw = 1.0f + fz, qx = fy, qy = -fx, qz = 0.0f;
    inv = 1.0f / sqrtf(qw * qw + qx * qx + qy * qy + qz * qz);
    qw *= inv; qx *= inv; qy *= inv; qz *= inv;

    // rot_mat = quat_to_rotmat(q) — every lane holds the full 3x3
    const float r00 = 1.0f - 2.0f * (qy * qy + qz * qz);
    const float r01 = 2.0f * (qx * qy - qw * qz);
    const float r02 = 2.0f * (qx * qz + qw * qy);
    const float r10 = 2.0f * (qx * qy + qw * qz);
    const float r11 = 1.0f - 2.0f * (qx * qx + qz * qz);
    const float r12 = 2.0f * (qy * qz - qw * qx);
    const float r20 = 2.0f * (qx * qz - qw * qy);
    const float r21 = 2.0f * (qy * qz + qw * qx);
    const float r22 = 1.0f - 2.0f * (qx * qx + qy * qy);

    // ---- WMMA einsum: D(16x16) = A(16x4: rot_mat rows, zero-padded)
    //                             x B(4x16: joint column vectors, zero-padded)
    // A layout (32-bit 16x4): lanes 0-15 rows M=0..15 {K0,K1}, lanes 16-31 {K2,K3}
    const bool hi = lane >= 16;
    const int  mn = lane & 15;  // M row (for A) / N col (for B/D)
    const float ra0 = hi ? r02 : r00, rb0 = hi ? r12 : r10, rc0 = hi ? r22 : r20;
    const float ra1 = hi ? 0.0f : r01, rb1 = hi ? 0.0f : r11, rc1 = hi ? 0.0f : r21;
    v2f A;
    A.x = (mn == 0) ? ra0 : (mn == 1) ? rb0 : (mn == 2) ? rc0 : 0.0f;
    A.y = (mn == 0) ? ra1 : (mn == 1) ? rb1 : (mn == 2) ? rc1 : 0.0f;

    // B layout (32-bit 4x16): lanes 0-15 {K0,K1}=(x,y), lanes 16-31 {K2,K3}=(z,0)
    // All shuffles unconditional (full EXEC), then lane-select.
    const int srcA = mn;        // groups 0,2
    const int srcB = mn + 16;   // groups 1,3
    v2f B0, B1, B2, B3;
    {
        float sx = shfl32(p0x, srcA), sy = shfl32(p0y, srcA), sz = shfl32(p0z, srcA);
        B0.x = hi ? sz : sx;  B0.y = hi ? 0.0f : sy;
    }
    {
        float sx = shfl32(p0x, srcB), sy = shfl32(p0y, srcB), sz = shfl32(p0z, srcB);
        B1.x = hi ? sz : sx;  B1.y = hi ? 0.0f : sy;
    }
    {
        float sx = shfl32(p1x, srcA), sy = shfl32(p1y, srcA), sz = shfl32(p1z, srcA);
        B2.x = hi ? sz : sx;  B2.y = hi ? 0.0f : sy;
    }
    {
        float sx = shfl32(p1x, srcB), sy = shfl32(p1y, srcB), sz = shfl32(p1z, srcB);
        B3.x = hi ? sz : sx;  B3.y = hi ? 0.0f : sy;
    }

    v8f cz = {};
    // D[i][v]: rows 0..2 = rotated coords, cols = 16 joints per group
    v8f D0 = __builtin_amdgcn_wmma_f32_16x16x4_f32(false, A, false, B0, (short)0, cz, false, false);
    v8f D1 = __builtin_amdgcn_wmma_f32_16x16x4_f32(false, A, false, B1, (short)0, cz, false, false);
    v8f D2 = __builtin_amdgcn_wmma_f32_16x16x4_f32(false, A, false, B2, (short)0, cz, false, false);
    v8f D3 = __builtin_amdgcn_wmma_f32_16x16x4_f32(false, A, false, B3, (short)0, cz, false, false);

    // Recombine so all 32 lanes store contiguous float3s.
    // IMPORTANT: shuffles issued unconditionally (convergent, full EXEC),
    // selection afterwards via cndmask — no divergent bpermute.
    const float s10 = shfl32(D1[0], mn);
    const float s11 = shfl32(D1[1], mn);
    const float s12 = shfl32(D1[2], mn);
    const float s30 = shfl32(D3[0], mn);
    const float s31 = shfl32(D3[1], mn);
    const float s32 = shfl32(D3[2], mn);
    {
        const float ox = hi ? s10 : D0[0];
        const float oy = hi ? s11 : D0[1];
        const float oz = hi ? s12 : D0[2];
        float* op = out_pos + (size_t)frame * 192 + lane * 3;
        op[0] = ox; op[1] = oy; op[2] = oz;
    }
    {
        const float ox = hi ? s30 : D2[0];
        const float oy = hi ? s31 : D2[1];
        const float oz = hi ? s32 : D2[2];
        float* op = out_pos + (size_t)frame * 192 + (lane + 32) * 3;
        op[0] = ox; op[1] = oy; op[2] = oz;
    }

    // ---- per-joint rotation -> axis-angle (root pre-multiplied by rot_mat) ----
    {
        float m[9];
#pragma unroll
        for (int k = 0; k < 9; ++k) m[k] = rf[v0 * 9 + k];
        // new_root = rot_mat @ m (only joint 0)
        float nm[9];
#pragma unroll
        for (int k = 0; k < 3; ++k) {
            nm[0 + k] = r00 * m[k] + r01 * m[3 + k] + r02 * m[6 + k];
            nm[3 + k] = r10 * m[k] + r11 * m[3 + k] + r12 * m[6 + k];
            nm[6 + k] = r20 * m[k] + r21 * m[3 + k] + r22 * m[6 + k];
        }
        const bool isroot = (v0 == 0);
#pragma unroll
        for (int k = 0; k < 9; ++k) m[k] = isroot ? nm[k] : m[k];
        float aa[3];
        mat_to_axis_angle(m, aa);
        float* op = out_aa + (size_t)frame * 192 + v0 * 3;
        op[0] = aa[0]; op[1] = aa[1]; op[2] = aa[2];
    }
    {
        float m[9];
#pragma unroll
        for (int k = 0; k < 9; ++k) m[k] = rf[v1 * 9 + k];
        float aa[3];
        mat_to_axis_angle(m, aa);
        float* op = out_aa + (size_t)frame * 192 + v1 * 3;
        op[0] = aa[0]; op[1] = aa[1]; op[2] = aa[2];
    }

    // ---- rot_mat output: 9 floats per frame, lanes 0..8 coalesced ----
    const float rmv =
        (lane == 0) ? r00 : (lane == 1) ? r01 : (lane == 2) ? r02 :
        (lane == 3) ? r10 : (lane == 4) ? r11 : (lane == 5) ? r12 :
        (lane == 6) ? r20 : (lane == 7) ? r21 : r22;
    if (lane < 9) out_rm[(size_t)frame * 9 + lane] = rmv;
}

extern "C" void kernel_launch(void* const* d_in, const int* in_sizes, int n_in,
                              void* d_out, int out_size, void* d_ws, size_t ws_size,
                              hipStream_t stream) {
    const float* rot = (const float*)d_in[0];   // rotations (B,F,V,3,3)
    const float* pos = (const float*)d_in[1];   // positions (B,F,V,3)
    float* out = (float*)d_out;

    const int npos    = in_sizes[1];     // B*F*V*3
    const int nframes = npos / 192;      // V=64, 3 components

    float* out_pos = out;                       // (B,F,V,3)
    float* out_aa  = out + (size_t)npos;        // (B,F,V,3)
    float* out_rm  = out + (size_t)npos * 2;    // (B,F,3,3)

    const int blocks = (nframes + 7) / 8;       // 8 waves (frames) per block
    facez_kernel<<<blocks, 256, 0, stream>>>(rot, pos, out_pos, out_aa, out_rm,
                                             nframes);
}